// NonMaximaSuppression2d_60687887893200
// MI455X (gfx1250) — compile-verified
//
#include <hip/hip_runtime.h>
#include <stdint.h>

// NMS2d (kornia NonMaximaSuppression2d) on (8,128,256,256) fp32.
// Bandwidth-bound: ~537 MB total traffic -> ~23us floor at 23.3 TB/s.
// Strategy: TDM (tensor_load_to_lds) DMA of a 34-row strip into LDS,
// 3x3 max stencil with replicate padding via index clamping,
// non-temporal b128 output stores (native vector type for the builtin).

typedef unsigned int v4u __attribute__((ext_vector_type(4)));
typedef int          v8i __attribute__((ext_vector_type(8)));
typedef int          v4i __attribute__((ext_vector_type(4)));
typedef float        v4f __attribute__((ext_vector_type(4)));

#define IMG_H 256
#define IMG_W 256
#define H_TILE 32                 // output rows per block
#define MAX_ROWS (H_TILE + 2)     // + top/bottom halo -> 34 KB LDS, ~9 blocks/WGP

__global__ __launch_bounds__(256)
void nms2d_tdm_kernel(const float* __restrict__ x, float* __restrict__ out) {
    __shared__ float tile[MAX_ROWS * IMG_W];   // 34 KB

    const int tiles_per_img = IMG_H / H_TILE;  // 8
    const int blk = blockIdx.x;
    const int bc  = blk / tiles_per_img;       // (b*C + c) plane index, 0..1023
    const int th  = blk % tiles_per_img;

    const int row_start = th * H_TILE;
    const int row_lo = (row_start - 1 < 0) ? 0 : (row_start - 1);
    const int row_hi = (row_start + H_TILE > IMG_H - 1) ? (IMG_H - 1)
                                                        : (row_start + H_TILE);
    const int nrows  = row_hi - row_lo + 1;    // 33 at edges, 34 interior

    const float* img = x + (size_t)bc * IMG_H * IMG_W;

    // ---- Wave 0 issues the TDM bulk load: global strip -> LDS ----
    if (threadIdx.x < 32) {
        const uint64_t gaddr = (uint64_t)(uintptr_t)(img + (size_t)row_lo * IMG_W);
        const uint32_t ldsa  = (uint32_t)(uintptr_t)&tile[0]; // low 32b = LDS byte offset

        // D# group 0 (128b): count=1 | lds_addr | global_addr[56:0] | type=2
        v4u g0;
        g0[0] = 1u;                                     // count=1 (valid), user mode
        g0[1] = ldsa;                                   // lds_addr (bytes)
        g0[2] = (uint32_t)gaddr;                        // global_addr[31:0]
        g0[3] = ((uint32_t)(gaddr >> 32) & 0x01FFFFFFu) // global_addr[56:32]
              | (2u << 30);                             // type = 2 ("image")

        // D# group 1 (256b): 2-D tensor, 4-byte elements, contiguous rows.
        v8i g1;
        g1[0] = (int)(2u << 16);                        // workgroup_mask=0, data_size=4B
        g1[1] = (int)(((uint32_t)IMG_W & 0xFFFFu) << 16);       // tensor_dim0 lo16 (=256)
        g1[2] = (int)((((uint32_t)IMG_W >> 16) & 0xFFFFu)       // tensor_dim0 hi16
              | (((uint32_t)nrows & 0xFFFFu) << 16));           // tensor_dim1 lo16
        g1[3] = (int)((((uint32_t)nrows >> 16) & 0xFFFFu)       // tensor_dim1 hi16
              | (((uint32_t)IMG_W & 0xFFFFu) << 16));           // tile_dim0 = 256
        g1[4] = (int)((uint32_t)nrows & 0xFFFFu);               // tile_dim1 = nrows, tile_dim2=0
        g1[5] = (int)(uint32_t)IMG_W;                           // tensor_dim0_stride lo32 (=256)
        g1[6] = 0;                                              // stride hi16 | dim1_stride lo16
        g1[7] = 0;

        v4i g2 = {0, 0, 0, 0};    // dims 2..3 unused (tile_dim2/3 = 0)
        v4i g3 = {0, 0, 0, 0};

#if defined(__clang_major__) && (__clang_major__ >= 23)
        v8i g4 = {0, 0, 0, 0, 0, 0, 0, 0};
        __builtin_amdgcn_tensor_load_to_lds(g0, g1, g2, g3, g4, 0);
#else
        __builtin_amdgcn_tensor_load_to_lds(g0, g1, g2, g3, 0);
#endif
        __builtin_amdgcn_s_wait_tensorcnt(0);
    }
    __syncthreads();   // DMA complete + visible to all 8 waves

    // ---- 3x3 NMS out of LDS; replicate padding by index clamping ----
    const int c0    = (threadIdx.x & 63) * 4;  // 64 col-groups of 4 cover W=256
    const int row_q = threadIdx.x >> 6;        // 4 rows processed per iteration

    // Clamped window columns are iteration-invariant; hoist them.
    int ccol[6];
    #pragma unroll
    for (int k = 0; k < 6; ++k) {
        int c = c0 - 1 + k;                              // replicate left/right
        ccol[k] = (c < 0) ? 0 : ((c > IMG_W - 1) ? (IMG_W - 1) : c);
    }

    float* outimg = out + (size_t)bc * IMG_H * IMG_W;

    #pragma unroll
    for (int it = 0; it < H_TILE / 4; ++it) {
        const int r  = row_start + it * 4 + row_q;           // global row
        const int rm = (r - 1 < 0) ? 0 : (r - 1);            // replicate top
        const int rp = (r + 1 > IMG_H - 1) ? (IMG_H - 1) : (r + 1);

        const float* Rm = &tile[(rm - row_lo) * IMG_W];
        const float* R0 = &tile[(r  - row_lo) * IMG_W];
        const float* Rp = &tile[(rp - row_lo) * IMG_W];

        float am[6], a0[6], ap[6];
        #pragma unroll
        for (int k = 0; k < 6; ++k) {
            am[k] = Rm[ccol[k]];
            a0[k] = R0[ccol[k]];
            ap[k] = Rp[ccol[k]];
        }

        v4f o;
        #pragma unroll
        for (int j = 0; j < 4; ++j) {
            const float xc = a0[j + 1];
            // Running max seeded at 0 == kornia's zeroed center conv channel.
            float m = 0.0f;
            m = fmaxf(m, am[j]); m = fmaxf(m, am[j + 1]); m = fmaxf(m, am[j + 2]);
            m = fmaxf(m, a0[j]);                           m = fmaxf(m, a0[j + 2]);
            m = fmaxf(m, ap[j]); m = fmaxf(m, ap[j + 1]); m = fmaxf(m, ap[j + 2]);
            o[j] = (xc > m) ? xc : 0.0f;
        }

        // Output is write-once streaming data: non-temporal b128 store keeps
        // the 192MB L2 for the TDM read stream + halo reuse instead.
        __builtin_nontemporal_store(o, (v4f*)(outimg + (size_t)r * IMG_W + c0));
    }
}

extern "C" void kernel_launch(void* const* d_in, const int* in_sizes, int n_in,
                              void* d_out, int out_size, void* d_ws, size_t ws_size,
                              hipStream_t stream) {
    const float* x  = (const float*)d_in[0];
    float* out      = (float*)d_out;

    const int BC    = 8 * 128;                       // batch * channels planes
    const int grid  = BC * (IMG_H / H_TILE);         // 8192 blocks
    nms2d_tdm_kernel<<<grid, 256, 0, stream>>>(x, out);
}